// UHGGraphSAGELayer_21328807592546
// MI455X (gfx1250) — compile-verified
//
#include <hip/hip_runtime.h>
#include <math.h>

#define FIN 128
#define FOUT 128
#define PROJ_EPS 1e-9f
#define LN_EPS 1e-5f

typedef float v2f __attribute__((ext_vector_type(2)));
typedef float v8f __attribute__((ext_vector_type(8)));

// ---------------------------------------------------------------- helpers
__device__ __forceinline__ void atomic_max_f32(float* addr, float val) {
    unsigned int* ua = reinterpret_cast<unsigned int*>(addr);
    unsigned int old = __float_as_uint(*addr);
    while (__uint_as_float(old) < val) {
        unsigned int assumed = old;
        old = atomicCAS(ua, assumed, __float_as_uint(val));
        if (old == assumed) break;
    }
}

// ---------------------------------------------------------------- K0: init scratch
__global__ void init_kernel(float* seg_max, float* denom, float* agg, int N) {
    long i = (long)blockIdx.x * blockDim.x + threadIdx.x;
    long total = (long)N * FIN;
    if (i < total) agg[i] = 0.0f;
    if (i < N) {
        seg_max[i] = -__builtin_inff();
        denom[i]   = 0.0f;
    }
}

// ---------------------------------------------------------------- K1: per-target max
__global__ void edge_max_kernel(const long long* ei, const float* w, float* seg_max, int E) {
    int e = blockIdx.x * blockDim.x + threadIdx.x;
    if (e >= E) return;
    int t = (int)ei[(long)E + e];   // edge_index[1][e] = target
    atomic_max_f32(&seg_max[t], w[e]);
}

// ---------------------------------------------------------------- K2: exp + denom
__global__ void edge_exp_kernel(const long long* ei, const float* w, const float* seg_max,
                                float* ew, float* denom, int E) {
    int e = blockIdx.x * blockDim.x + threadIdx.x;
    if (e >= E) return;
    int t = (int)ei[(long)E + e];
    float v = __expf(w[e] - seg_max[t]);
    ew[e] = v;
    atomicAdd(&denom[t], v);
}

// ---------------------------------------------------------------- K3: weighted scatter
// one wave32 per edge, each lane owns 4 consecutive features (float4 gather,
// 4 x global_atomic_add_f32 scatter). x and agg both fit in the 192MB L2.
__global__ void edge_scatter_kernel(const long long* ei, const float* ew, const float* denom,
                                    const float* x, float* agg, int E) {
    int gid  = blockIdx.x * blockDim.x + threadIdx.x;
    int e    = gid >> 5;
    int lane = threadIdx.x & 31;
    if (e >= E) return;
    int s = (int)ei[e];
    int t = (int)ei[(long)E + e];
    float alpha = ew[e] / denom[t];
    const float4* xs = (const float4*)(x + (long)s * FIN);
    float4 v = xs[lane];
    float* dst = agg + (long)t * FIN + lane * 4;
    atomicAdd(dst + 0, alpha * v.x);
    atomicAdd(dst + 1, alpha * v.y);
    atomicAdd(dst + 2, alpha * v.z);
    atomicAdd(dst + 3, alpha * v.w);
}

// ---------------------------------------------------------------- K4: fused dual-GEMM
// One wave owns a 16x128 output block of S = x@Ws + agg_n@Wn + (bs+bn).
// K-outer loop: A fragments loaded once per k-step (8x fewer than tile-outer).
// Both weight matrices are staged in LDS in a k-pair layout so each B fragment
// (W[k][n], W[k+1][n]) is a single ds_load_b64; odd pair-rows are rotated by
// 32 dwords so lo/hi half-wave reads hit disjoint LDS bank groups.
// Projective normalize + layernorm fold into per-row scalars computed with
// half-wave shfl_xor reductions (D rows live in fixed lane halves).
__global__ void __launch_bounds__(256)
fused_gemm_kernel(const float* __restrict__ x, const float* __restrict__ agg,
                  const float* __restrict__ Ws, const float* __restrict__ bs,
                  const float* __restrict__ Wn, const float* __restrict__ bn,
                  const float* __restrict__ gamma, const float* __restrict__ beta,
                  float* __restrict__ out, int N) {
    __shared__ float lw[2 * FIN * FOUT];          // 128 KB: Ws | Wn, pair-swizzled
    float* lws = lw;
    float* lwn = lw + FIN * FOUT;

    // ---- stage both weight matrices (coalesced global reads)
    for (int i = threadIdx.x; i < FIN * FOUT; i += 256) {
        const int k = i >> 7, n = i & 127;
        const int row = k >> 1;                                   // k-pair row
        const int col = (((n << 1) + ((row & 1) << 5)) & 255) + (k & 1);
        const int a = (row << 8) + col;
        lws[a] = Ws[i];
        lwn[a] = Wn[i];
    }
    __syncthreads();

    const int lane = threadIdx.x & 31;
    const int rb = blockIdx.x * (blockDim.x >> 5) + (threadIdx.x >> 5);
    const long r0 = (long)rb * 16;
    if (r0 >= N) return;

    const int m  = lane & 15;    // row (A) / col (B,D) index within tile
    const int hi = lane >> 4;    // half-wave select
    const int kb = hi << 1;      // K sub-offset for 16x4 A / 4x16 B fragments

    const float* aggRow = agg + (r0 + m) * FIN;
    const float* xRow   = x   + (r0 + m) * FIN;

    // --- L2 norm of agg row (r0+m): lanes l and l+16 jointly cover all 128 k's
    float ssq = 0.0f;
    for (int k0 = 0; k0 < FIN; k0 += 4) {
        v2f a = *(const v2f*)(aggRow + k0 + kb);
        ssq += a.x * a.x + a.y * a.y;
    }
    ssq += __shfl_xor(ssq, 16, 32);
    const float rnorm = 1.0f / (sqrtf(ssq) + PROJ_EPS);

    // --- init accumulators with bias, accumulate over K with all 8 tiles live
    v8f acc[8];
    #pragma unroll
    for (int t = 0; t < 8; ++t) {
        const float bias = bs[t * 16 + m] + bn[t * 16 + m];
        #pragma unroll
        for (int r = 0; r < 8; ++r) acc[t][r] = bias;
    }

    for (int k0 = 0; k0 < FIN; k0 += 4) {
        const int kk = k0 + kb;                     // even
        v2f a1 = *(const v2f*)(xRow + kk);
        v2f a2 = *(const v2f*)(aggRow + kk);
        a2.x *= rnorm; a2.y *= rnorm;               // fold agg row-normalize
        const int prow = kk >> 1;
        const int pbase = prow << 8;
        const int shift = (prow & 1) << 5;          // bank de-conflict rotation
        #pragma unroll
        for (int t = 0; t < 8; ++t) {
            const int col = ((t << 5) + (m << 1) + shift) & 255;
            v2f b1 = *(const v2f*)(lws + pbase + col);
            acc[t] = __builtin_amdgcn_wmma_f32_16x16x4_f32(false, a1, false, b1,
                                                           (short)0, acc[t], false, false);
            v2f b2 = *(const v2f*)(lwn + pbase + col);
            acc[t] = __builtin_amdgcn_wmma_f32_16x16x4_f32(false, a2, false, b2,
                                                           (short)0, acc[t], false, false);
        }
    }

    // --- per-row stats. comb = 0.5*S / (0.5*sqrt(sum S^2) + eps) = S * inv,
    //     inv = 1/(sqrt(s2) + 2*eps); layernorm folds into one affine per row.
    #pragma unroll
    for (int r = 0; r < 8; ++r) {
        float s1 = 0.0f, s2 = 0.0f;
        #pragma unroll
        for (int t = 0; t < 8; ++t) {
            float v = acc[t][r];
            s1 += v;
            s2 += v * v;
        }
        #pragma unroll
        for (int d = 1; d < 16; d <<= 1) {
            s1 += __shfl_xor(s1, d, 32);
            s2 += __shfl_xor(s2, d, 32);
        }
        const float mu   = s1 * (1.0f / 128.0f);
        const float varS = s2 * (1.0f / 128.0f) - mu * mu;
        const float inv  = 1.0f / (sqrtf(s2) + 2.0f * PROJ_EPS);
        const float scl  = inv * rsqrtf(inv * inv * varS + LN_EPS);
        const long row   = r0 + r + (hi << 3);
        #pragma unroll
        for (int t = 0; t < 8; ++t) {
            const int n = t * 16 + m;
            out[row * FOUT + n] = (acc[t][r] - mu) * scl * gamma[n] + beta[n];
        }
    }
}

// ---------------------------------------------------------------- launcher
extern "C" void kernel_launch(void* const* d_in, const int* in_sizes, int n_in,
                              void* d_out, int out_size, void* d_ws, size_t ws_size,
                              hipStream_t stream) {
    const float*     x     = (const float*)d_in[0];
    const long long* ei    = (const long long*)d_in[1];
    const float*     w     = (const float*)d_in[2];
    const float*     Ws    = (const float*)d_in[3];
    const float*     bs    = (const float*)d_in[4];
    const float*     Wn    = (const float*)d_in[5];
    const float*     bn    = (const float*)d_in[6];
    const float*     gamma = (const float*)d_in[7];
    const float*     beta  = (const float*)d_in[8];
    float* out = (float*)d_out;

    const int N = in_sizes[0] / FIN;
    const int E = in_sizes[2];

    // workspace layout (floats): agg[N*128] | seg_max[N] | denom[N] | ew[E]
    float* agg     = (float*)d_ws;
    float* seg_max = agg + (size_t)N * FIN;
    float* denom   = seg_max + N;
    float* ew      = denom + N;

    {   // K0: zero agg, init seg_max/denom
        long total = (long)N * FIN;
        int blocks = (int)((total + 255) / 256);
        init_kernel<<<blocks, 256, 0, stream>>>(seg_max, denom, agg, N);
    }
    {   // K1: per-target running max
        int blocks = (E + 255) / 256;
        edge_max_kernel<<<blocks, 256, 0, stream>>>(ei, w, seg_max, E);
    }
    {   // K2: exp + denominator
        int blocks = (E + 255) / 256;
        edge_exp_kernel<<<blocks, 256, 0, stream>>>(ei, w, seg_max, ew, denom, E);
    }
    {   // K3: weighted scatter, one wave32 per edge
        long threads = (long)E * 32;
        int blocks = (int)((threads + 255) / 256);
        edge_scatter_kernel<<<blocks, 256, 0, stream>>>(ei, ew, denom, x, agg, E);
    }
    {   // K4: fused WMMA dual-GEMM + projective norm + layernorm
        int rowblocks = (N + 15) / 16;
        int blocks = (rowblocks + 7) / 8;   // 8 waves per 256-thread block
        fused_gemm_kernel<<<blocks, 256, 0, stream>>>(x, agg, Ws, bs, Wn, bn,
                                                      gamma, beta, out, N);
    }
}